// NTM_71932112274166
// MI455X (gfx1250) — compile-verified
//
#include <hip/hip_runtime.h>
#include <hip/hip_bf16.h>
#include <math.h>

#define BB 512
#define NN 1024
#define MM 128
#define HH 512
#define NHH 4
#define EPSF 1e-16f

typedef float v2f __attribute__((ext_vector_type(2)));
typedef float v8f __attribute__((ext_vector_type(8)));

__device__ __forceinline__ float sigmoidf_(float x) { return 1.f / (1.f + expf(-x)); }
__device__ __forceinline__ float softplusf_(float x) { return (x > 20.f) ? x : log1pf(expf(x)); }

// ---------------------------------------------------------------------------
// Pack A = [x | prev_reads(transposed) | h0]  -> [512, 1152]
// ---------------------------------------------------------------------------
__global__ void ntm_pack_gates_in(const float* __restrict__ x,
                                  const float* __restrict__ pr,   // [NH,B,M]
                                  const float* __restrict__ h0,   // [B,H]
                                  float* __restrict__ inp) {
  int idx = blockIdx.x * 256 + threadIdx.x;           // < 512*1152
  int b = idx / 1152, c = idx % 1152;
  float v;
  if (c < 128) {
    v = x[b * 128 + c];
  } else if (c < 640) {
    int j = c - 128, hh = j >> 7, m = j & 127;
    v = pr[(hh * BB + b) * MM + m];
  } else {
    v = h0[b * HH + (c - 640)];
  }
  inp[idx] = v;
}

// Pack W = [W_ih | W_hh] -> [2048, 1152]
__global__ void ntm_pack_wcomb(const float* __restrict__ Wih,
                               const float* __restrict__ Whh,
                               float* __restrict__ Wc) {
  int idx = blockIdx.x * 256 + threadIdx.x;           // < 2048*1152
  int n = idx / 1152, k = idx % 1152;
  Wc[idx] = (k < 640) ? Wih[n * 640 + k] : Whh[n * HH + (k - 640)];
}

// ---------------------------------------------------------------------------
// C[M,N] = act(A[M,K] @ W[N,K]^T + bias + bias2), fp32 WMMA 16x16x4.
// block = 128 threads (4 waves). Wave tile 16(M) x 32(N): two accumulators
// share one A fragment. Block tile 32 x 64.
// Out-of-range columns: W row pointer is CLAMPED (valid memory, finite
// garbage) -- WMMA output column n only depends on B column n, so garbage
// lands exclusively in columns that the bounds-checked store discards.
// Requires Mdim % 32 == 0, Kdim % 32 == 0 (true for all uses here).
// ---------------------------------------------------------------------------
__global__ void ntm_gemm_wmma_f32(const float* __restrict__ A,
                                  const float* __restrict__ W,
                                  const float* __restrict__ bias,
                                  const float* __restrict__ bias2,
                                  float* __restrict__ C,
                                  int Ndim, int Kdim, int act) {
  const int lane = threadIdx.x & 31;
  const int wave = threadIdx.x >> 5;
  const int row0 = blockIdx.y * 32 + (wave >> 1) * 16;
  const int col0 = blockIdx.x * 64 + (wave & 1) * 32;
  const int mi = lane & 15;
  const int kg = (lane >> 4) << 1;                    // 0 or 2
  const float* Aptr = A + (size_t)(row0 + mi) * Kdim + kg;
  int bc0 = col0 + mi;       if (bc0 > Ndim - 1) bc0 = Ndim - 1;
  int bc1 = col0 + 16 + mi;  if (bc1 > Ndim - 1) bc1 = Ndim - 1;
  const float* Wptr0 = W + (size_t)bc0 * Kdim + kg;
  const float* Wptr1 = W + (size_t)bc1 * Kdim + kg;

  v8f acc0 = {0.f, 0.f, 0.f, 0.f, 0.f, 0.f, 0.f, 0.f};
  v8f acc1 = {0.f, 0.f, 0.f, 0.f, 0.f, 0.f, 0.f, 0.f};
#pragma unroll 8
  for (int k = 0; k < Kdim; k += 4) {
    v2f af = *(const v2f*)(Aptr + k);
    v2f bf0 = *(const v2f*)(Wptr0 + k);
    v2f bf1 = *(const v2f*)(Wptr1 + k);
    acc0 = __builtin_amdgcn_wmma_f32_16x16x4_f32(false, af, false, bf0,
                                                 (short)0, acc0, false, false);
    acc1 = __builtin_amdgcn_wmma_f32_16x16x4_f32(false, af, false, bf1,
                                                 (short)0, acc1, false, false);
  }

  const int rbase = row0 + ((lane >> 4) << 3);
  const int cn0 = col0 + (lane & 15);
  const int cn1 = cn0 + 16;
  if (cn0 < Ndim) {
    float bv = (bias ? bias[cn0] : 0.f) + (bias2 ? bias2[cn0] : 0.f);
    for (int v = 0; v < 8; ++v) {
      float val = acc0[v] + bv;
      if (act == 1) val = sigmoidf_(val);
      C[(size_t)(rbase + v) * Ndim + cn0] = val;
    }
  }
  if (cn1 < Ndim) {
    float bv = (bias ? bias[cn1] : 0.f) + (bias2 ? bias2[cn1] : 0.f);
    for (int v = 0; v < 8; ++v) {
      float val = acc1[v] + bv;
      if (act == 1) val = sigmoidf_(val);
      C[(size_t)(rbase + v) * Ndim + cn1] = val;
    }
  }
}

// ---------------------------------------------------------------------------
// LSTM cell elementwise: gates [B, 4H] split [i f g o]
// ---------------------------------------------------------------------------
__global__ void ntm_lstm(const float* __restrict__ gates,
                         const float* __restrict__ c0,
                         float* __restrict__ h_out, float* __restrict__ c_out) {
  int idx = blockIdx.x * 256 + threadIdx.x;           // < B*H
  int b = idx >> 9, j = idx & 511;
  size_t base = (size_t)b * 2048 + j;
  float ig = gates[base], fg = gates[base + 512];
  float gg = gates[base + 1024], og = gates[base + 1536];
  float c = sigmoidf_(fg) * c0[idx] + sigmoidf_(ig) * tanhf(gg);
  float h = sigmoidf_(og) * tanhf(c);
  h_out[idx] = h;
  c_out[idx] = c;
}

__global__ void ntm_zero(float* __restrict__ p, int n) {
  int idx = blockIdx.x * 256 + threadIdx.x;
  if (idx < n) p[idx] = 0.f;
}

// ---------------------------------------------------------------------------
// Per (b,n) row: dot_r = mem.k_r, dot_w = mem.k_w, norm = ||mem row||
// one wave per row, 8 rows per 256-thread block
// ---------------------------------------------------------------------------
__global__ void ntm_dots(const float* __restrict__ mem,
                         const float* __restrict__ p,   // [B,134], k=cols 0..127
                         const float* __restrict__ q,   // [B,390]
                         float* __restrict__ dot_r, float* __restrict__ dot_w,
                         float* __restrict__ norms) {
  const int lane = threadIdx.x & 31;
  const int wave = threadIdx.x >> 5;
  const size_t row = (size_t)blockIdx.x * 8 + wave;    // b*N + n
  const int b = (int)(row >> 10);
  const float4 mv = ((const float4*)(mem + row * MM))[lane];
  const float2* kr2 = (const float2*)(p + (size_t)b * 134);
  const float2* kw2 = (const float2*)(q + (size_t)b * 390);
  float2 kr0 = kr2[lane * 2], kr1 = kr2[lane * 2 + 1];
  float2 kw0 = kw2[lane * 2], kw1 = kw2[lane * 2 + 1];
  float dr = mv.x * kr0.x + mv.y * kr0.y + mv.z * kr1.x + mv.w * kr1.y;
  float dw = mv.x * kw0.x + mv.y * kw0.y + mv.z * kw1.x + mv.w * kw1.y;
  float ss = mv.x * mv.x + mv.y * mv.y + mv.z * mv.z + mv.w * mv.w;
  for (int off = 16; off > 0; off >>= 1) {
    dr += __shfl_xor(dr, off, 32);
    dw += __shfl_xor(dw, off, 32);
    ss += __shfl_xor(ss, off, 32);
  }
  if (lane == 0) {
    dot_r[row] = dr;
    dot_w[row] = dw;
    norms[row] = sqrtf(ss);
  }
}

// ---------------------------------------------------------------------------
// Addressing: cosine softmax -> interpolate -> circular shift -> sharpen ->
// normalize. grid (B, 2): y==0 read head, y==1 write head. block = 256.
// ---------------------------------------------------------------------------
__device__ __forceinline__ float blk_sum(float v, float* red) {
  int tid = threadIdx.x;
  red[tid] = v; __syncthreads();
  for (int s = 128; s > 0; s >>= 1) { if (tid < s) red[tid] += red[tid + s]; __syncthreads(); }
  float r = red[0]; __syncthreads();
  return r;
}
__device__ __forceinline__ float blk_max(float v, float* red) {
  int tid = threadIdx.x;
  red[tid] = v; __syncthreads();
  for (int s = 128; s > 0; s >>= 1) { if (tid < s) red[tid] = fmaxf(red[tid], red[tid + s]); __syncthreads(); }
  float r = red[0]; __syncthreads();
  return r;
}

__global__ void ntm_address(const float* __restrict__ p, const float* __restrict__ q,
                            const float* __restrict__ dot_r, const float* __restrict__ dot_w,
                            const float* __restrict__ norms,
                            const float* __restrict__ prev_w,   // [2*NH, B, N]
                            float* __restrict__ ws_out,         // [2*NH, B, N]
                            int head) {
  __shared__ float red[256];
  __shared__ float sval[NN];
  __shared__ float swg[NN];
  const int b = blockIdx.x;
  const int rw = blockIdx.y;                           // 0 read, 1 write
  const int tid = threadIdx.x;
  const float* P = rw ? (q + (size_t)b * 390) : (p + (size_t)b * 134);
  const float* dotP = (rw ? dot_w : dot_r) + (size_t)b * NN;
  const float* nrm = norms + (size_t)b * NN;
  const int slot = 2 * head + rw;
  const float* pw = prev_w + ((size_t)slot * BB + b) * NN;
  float* wout = ws_out + ((size_t)slot * BB + b) * NN;

  // scalar params
  float beta = softplusf_(P[128]);
  float g = sigmoidf_(P[129]);
  float s0r = P[130], s1r = P[131], s2r = P[132];
  float sm = fmaxf(s0r, fmaxf(s1r, s2r));
  float e0 = expf(s0r - sm), e1 = expf(s1r - sm), e2 = expf(s2r - sm);
  float esum = e0 + e1 + e2;
  float s0 = e0 / esum, s1 = e1 / esum, s2 = e2 / esum;
  float gamma = 1.f + softplusf_(P[133]);

  // ||k||
  float kk = 0.f;
  if (tid < 128) { float kv = P[tid]; kk = kv * kv; }
  float knorm = fmaxf(sqrtf(blk_sum(kk, red)), EPSF);

  // content scores + softmax
  float lmax = -INFINITY;
  for (int n = tid; n < NN; n += 256) {
    float v = beta * dotP[n] / (fmaxf(nrm[n], EPSF) * knorm);
    sval[n] = v;
    lmax = fmaxf(lmax, v);
  }
  float vmax = blk_max(lmax, red);
  float lsum = 0.f;
  for (int n = tid; n < NN; n += 256) {
    float e = expf(sval[n] - vmax);
    sval[n] = e;
    lsum += e;
  }
  float inv = 1.f / blk_sum(lsum, red);

  // interpolate
  for (int n = tid; n < NN; n += 256)
    swg[n] = g * (sval[n] * inv) + (1.f - g) * pw[n];
  __syncthreads();

  // shift + sharpen
  float lps = 0.f;
  for (int n = tid; n < NN; n += 256) {
    float wh = s0 * swg[(n + NN - 1) & (NN - 1)] + s1 * swg[n] + s2 * swg[(n + 1) & (NN - 1)];
    float w = powf(fmaxf(wh, 0.f), gamma);
    sval[n] = w;
    lps += w;
  }
  float invs = 1.f / (blk_sum(lps, red) + EPSF);
  for (int n = tid; n < NN; n += 256) wout[n] = sval[n] * invs;
}

// ---------------------------------------------------------------------------
// Fused read vector + erase/add memory update. grid (B, 8), block = 128.
// thread t = column, chunk = 128 rows of n.
// ---------------------------------------------------------------------------
__global__ void ntm_read_update(float* __restrict__ mem,
                                const float* __restrict__ q,   // [B,390]
                                const float* __restrict__ wsv, // new_ws [2NH,B,N]
                                float* __restrict__ reads_out, // [NH,B,M]
                                int head) {
  const int b = blockIdx.x;
  const int chunk = blockIdx.y;
  const int t = threadIdx.x;
  const float* wr = wsv + ((size_t)(2 * head) * BB + b) * NN;
  const float* ww = wr + (size_t)BB * NN;
  const float e = sigmoidf_(q[(size_t)b * 390 + 134 + t]);
  const float a = q[(size_t)b * 390 + 262 + t];
  const int n0 = chunk * 128;
  float* mcol = mem + ((size_t)b * NN + n0) * MM + t;
  float racc = 0.f;
  for (int nn = 0; nn < 128; ++nn) {
    __builtin_prefetch(mcol + (size_t)(nn + 4) * MM, 0, 0);
    float wrv = wr[n0 + nn];
    float wwv = ww[n0 + nn];
    float mv = mcol[(size_t)nn * MM];
    racc += wrv * mv;
    mcol[(size_t)nn * MM] = mv * (1.f - wwv * e) + wwv * a;
  }
  atomicAdd(reads_out + ((size_t)head * BB + b) * MM + t, racc);
}

// Pack inp2 = [ctrl | reads0..3]  -> [512, 1024]
__global__ void ntm_pack_inp2(const float* __restrict__ h_out,
                              const float* __restrict__ reads_out,
                              float* __restrict__ inp2) {
  int idx = blockIdx.x * 256 + threadIdx.x;           // < 512*1024
  int b = idx >> 10, j = idx & 1023;
  float v;
  if (j < 512) v = h_out[b * HH + j];
  else {
    int jj = j - 512;
    v = reads_out[((size_t)(jj >> 7) * BB + b) * MM + (jj & 127)];
  }
  inp2[idx] = v;
}

// ---------------------------------------------------------------------------
extern "C" void kernel_launch(void* const* d_in, const int* in_sizes, int n_in,
                              void* d_out, int out_size, void* d_ws, size_t ws_size,
                              hipStream_t stream) {
  const float* x         = (const float*)d_in[0];
  const float* prev_reads= (const float*)d_in[1];
  const float* h0        = (const float*)d_in[2];
  const float* c0        = (const float*)d_in[3];
  const float* prev_w    = (const float*)d_in[4];
  const float* prev_mem  = (const float*)d_in[5];
  const float* W_ih      = (const float*)d_in[6];
  const float* W_hh      = (const float*)d_in[7];
  const float* b_ih      = (const float*)d_in[8];
  const float* b_hh      = (const float*)d_in[9];
  const float* Wr        = (const float*)d_in[10];
  const float* br        = (const float*)d_in[11];
  const float* Ww        = (const float*)d_in[12];
  const float* bw        = (const float*)d_in[13];
  const float* fc_W      = (const float*)d_in[14];
  const float* fc_b      = (const float*)d_in[15];

  // output layout: o, mem, reads, new_ws, h, c
  float* out       = (float*)d_out;
  float* o_out     = out;                               // 512*128
  float* mem_out   = o_out + 512 * 128;                 // 512*1024*128
  float* reads_out = mem_out + (size_t)512 * 1024 * 128;// 4*512*128
  float* nws_out   = reads_out + 4 * 512 * 128;         // 8*512*1024
  float* h_out     = nws_out + (size_t)8 * 512 * 1024;  // 512*512
  float* c_out     = h_out + 512 * 512;                 // 512*512

  // workspace
  float* w       = (float*)d_ws;
  float* inpcat  = w;                    // 512*1152   = 589824
  float* Wcomb   = inpcat + 589824;      // 2048*1152  = 2359296
  float* gates   = Wcomb + 2359296;      // 512*2048   = 1048576
  float* pbuf    = gates + 1048576;      // 512*134    = 68608
  float* qbuf    = pbuf + 68608;         // 512*390    = 199680
  float* dot_r   = qbuf + 199680;        // 512*1024
  float* dot_w   = dot_r + 524288;
  float* nrm     = dot_w + 524288;
  float* inp2    = nrm + 524288;         // 512*1024

  // mem starts as copy of prev_mem (inputs must not be mutated)
  hipMemcpyAsync(mem_out, prev_mem, sizeof(float) * (size_t)512 * 1024 * 128,
                 hipMemcpyDeviceToDevice, stream);

  ntm_pack_gates_in<<<589824 / 256, 256, 0, stream>>>(x, prev_reads, h0, inpcat);
  ntm_pack_wcomb<<<2359296 / 256, 256, 0, stream>>>(W_ih, W_hh, Wcomb);

  // gates = inpcat @ Wcomb^T + b_ih + b_hh    [512 x 2048], K=1152
  ntm_gemm_wmma_f32<<<dim3(2048 / 64, 512 / 32), 128, 0, stream>>>(
      inpcat, Wcomb, b_ih, b_hh, gates, 2048, 1152, 0);

  ntm_lstm<<<(512 * 512) / 256, 256, 0, stream>>>(gates, c0, h_out, c_out);

  ntm_zero<<<(4 * 512 * 128) / 256, 256, 0, stream>>>(reads_out, 4 * 512 * 128);

  for (int i = 0; i < NHH; ++i) {
    // p = ctrl @ Wr[i]^T + br[i]   [512 x 134], K=512
    ntm_gemm_wmma_f32<<<dim3(3, 512 / 32), 128, 0, stream>>>(
        h_out, Wr + (size_t)i * 134 * 512, br + (size_t)i * 134, nullptr,
        pbuf, 134, 512, 0);
    // q = ctrl @ Ww[i]^T + bw[i]   [512 x 390], K=512
    ntm_gemm_wmma_f32<<<dim3(7, 512 / 32), 128, 0, stream>>>(
        h_out, Ww + (size_t)i * 390 * 512, bw + (size_t)i * 390, nullptr,
        qbuf, 390, 512, 0);

    ntm_dots<<<(512 * 1024) / 8, 256, 0, stream>>>(mem_out, pbuf, qbuf,
                                                   dot_r, dot_w, nrm);
    ntm_address<<<dim3(512, 2), 256, 0, stream>>>(pbuf, qbuf, dot_r, dot_w, nrm,
                                                  prev_w, nws_out, i);
    ntm_read_update<<<dim3(512, 8), 128, 0, stream>>>(mem_out, qbuf, nws_out,
                                                      reads_out, i);
  }

  ntm_pack_inp2<<<(512 * 1024) / 256, 256, 0, stream>>>(h_out, reads_out, inp2);

  // o = sigmoid(inp2 @ fc_W^T + fc_b)   [512 x 128], K=1024
  ntm_gemm_wmma_f32<<<dim3(128 / 64, 512 / 32), 128, 0, stream>>>(
      inp2, fc_W, fc_b, nullptr, o_out, 128, 1024, 1);
}